// CrossAttentionBlock_48730698941055
// MI455X (gfx1250) — compile-verified
//
#include <hip/hip_runtime.h>

typedef __attribute__((ext_vector_type(2))) float v2f;
typedef __attribute__((ext_vector_type(4))) float v4f;
typedef __attribute__((ext_vector_type(8))) float v8f;

#define B_    16
#define C_    512
#define CTX_  1024
#define HW_   4096
#define VSTRIDE 516   // 512 + 4 pad: (row*516 + k) % 64 = (4*row + k) % 64 -> conflict-free LDS

// ---------------------------------------------------------------------------
// Kernel A: fused tiny GEMMs, one workgroup of 1024 threads = 32 waves.
//   Stage 1: V  = context(16x1024) @ Wv^T(512x1024) + kv_b[512:]   -> LDS
//   Stage 2: PV = V(16x512)        @ proj_w^T(512x512) + proj_b    -> d_ws
// Each wave owns one 16-wide output column tile; K is marched in steps of 4
// with V_WMMA_F32_16X16X4_F32 (exact fp32).
// ---------------------------------------------------------------------------
__global__ __launch_bounds__(1024) void ca_small_gemms(
    const float* __restrict__ context,   // (16, 1024)
    const float* __restrict__ kv_w,      // (1024, 1024); rows 512.. = Wv
    const float* __restrict__ kv_b,      // (1024)
    const float* __restrict__ proj_w,    // (512, 512)
    const float* __restrict__ proj_b,    // (512)
    float* __restrict__ pv)              // out: (16, 512)
{
    __shared__ float vbuf[B_ * VSTRIDE];          // ~33 KB of 320 KB

    const int lane  = threadIdx.x & 31;
    const int wave  = threadIdx.x >> 5;           // 0..31 -> N-tile
    const int row   = lane & 15;                  // A: M row / B: N col
    const int khalf = (lane >> 4) << 1;           // K sub-offset 0 or 2
    const int mbase = (lane >> 4) << 3;           // C/D: M base 0 or 8

    // ---- Stage 1: V = context @ Wv^T + bias ----
    {
        const int N0 = wave * 16;
        v8f acc = {};
        const float* arow = context + (size_t)row * CTX_;                 // A[M=row, k]
        const float* brow = kv_w + (size_t)(C_ + N0 + row) * CTX_;        // B[k, N=N0+row]
        for (int k0 = 0; k0 < CTX_; k0 += 4) {
            const int kb = k0 + khalf;
            v2f a = *(const v2f*)(arow + kb);
            v2f b = *(const v2f*)(brow + kb);
            acc = __builtin_amdgcn_wmma_f32_16x16x4_f32(
                false, a, false, b, (short)0, acc, false, false);
        }
        const int n    = N0 + row;
        const float bs = kv_b[C_ + n];
#pragma unroll
        for (int r = 0; r < 8; ++r)
            vbuf[(mbase + r) * VSTRIDE + n] = acc[r] + bs;
    }
    __syncthreads();

    // ---- Stage 2: PV = V @ proj_w^T + proj_b ----
    {
        const int O0 = wave * 16;
        v8f acc = {};
        const float* arow = vbuf + (size_t)row * VSTRIDE;                 // A from LDS
        const float* brow = proj_w + (size_t)(O0 + row) * C_;             // B[k, N=O0+row]
        for (int k0 = 0; k0 < C_; k0 += 4) {
            const int kb = k0 + khalf;
            v2f a = *(const v2f*)(arow + kb);
            v2f b = *(const v2f*)(brow + kb);
            acc = __builtin_amdgcn_wmma_f32_16x16x4_f32(
                false, a, false, b, (short)0, acc, false, false);
        }
        const int o    = O0 + row;
        const float bs = proj_b[o];
#pragma unroll
        for (int r = 0; r < 8; ++r)
            pv[(size_t)(mbase + r) * C_ + o] = acc[r] + bs;
    }
}

// ---------------------------------------------------------------------------
// Kernel B: out = x + pv[b,c] broadcast over HW. Pure streaming: 256 MB total
// traffic -> ~11.5 us at 23.3 TB/s. float4 (GLOBAL_LOAD/STORE_B128), exact grid
// (no predication). 1024 float4 per (b,c) row, so pv index = i4 >> 10 and is
// uniform per 256-thread block (broadcast scalar-cached load).
// ---------------------------------------------------------------------------
__global__ __launch_bounds__(256) void ca_residual_add(
    const float* __restrict__ x,
    const float* __restrict__ pv,
    float* __restrict__ out)
{
    const size_t i4  = (size_t)blockIdx.x * 256 + threadIdx.x;
    const float  add = pv[i4 >> 10];
    const v4f xv = ((const v4f*)x)[i4];
    v4f o;
    o.x = xv.x + add; o.y = xv.y + add; o.z = xv.z + add; o.w = xv.w + add;
    ((v4f*)out)[i4] = o;
}

extern "C" void kernel_launch(void* const* d_in, const int* in_sizes, int n_in,
                              void* d_out, int out_size, void* d_ws, size_t ws_size,
                              hipStream_t stream) {
    // setup_inputs order:
    // 0:x 1:context 2:norm_w 3:norm_b 4:q_w 5:q_b 6:kv_w 7:kv_b 8:proj_w 9:proj_b
    const float* x       = (const float*)d_in[0];
    const float* context = (const float*)d_in[1];
    const float* kv_w    = (const float*)d_in[6];
    const float* kv_b    = (const float*)d_in[7];
    const float* proj_w  = (const float*)d_in[8];
    const float* proj_b  = (const float*)d_in[9];
    float* out = (float*)d_out;
    float* pv  = (float*)d_ws;                    // 16*512 floats = 32 KB scratch

    ca_small_gemms<<<1, 1024, 0, stream>>>(context, kv_w, kv_b, proj_w, proj_b, pv);

    const size_t total_f4 = (size_t)B_ * C_ * HW_ / 4;   // 8,388,608
    ca_residual_add<<<(int)(total_f4 / 256), 256, 0, stream>>>(x, pv, out);
}